// DeformConv3D_11948599017925
// MI455X (gfx1250) — compile-verified
//
#include <hip/hip_runtime.h>
#include <math.h>

// ---------------- problem constants (from reference) ----------------
#define Bn   2
#define Dn   8
#define Nn   16      // B*D images
#define Cin  128
#define OG   4       // offset groups
#define CG   32      // Cin/OG
#define OC   128     // out channels
#define OPG  32      // OC/OG
#define Hh   64
#define Ww   64
#define HW   4096
#define KK   9       // 3x3 taps
#define OFFC 72      // 2*KK*OG offset channels
#define RED  288     // CG*KK reduction length (one channel-chunk)
#define SPITCH 17    // padded LDS row pitch (avoid bank conflicts)

typedef float v2f __attribute__((ext_vector_type(2)));
typedef float v8f __attribute__((ext_vector_type(8)));

// ---------------------------------------------------------------
// Kernel 1: offset conv (3x3, pad 1, 128 -> 72 ch) as implicit-GEMM
// on fp32 WMMA.  Each wave: one 16-pixel tile of one image.
//   D[80x16] (5 M-tiles, rows >= 72 padded) = W[80 x 1152] * im2col.
//   K = 1152 split into 4 chunks of 288; LDS tile rebuilt per chunk.
// Padded A rows are CLAMPED (not zeroed): row M of D only depends on
// row M of A, and rows 72..79 of D are never stored, so clamping the
// address keeps the inner loop divergence-free (no saveexec).
// ---------------------------------------------------------------
__global__ void offset_wmma_kernel(const float* __restrict__ x,
                                   const float* __restrict__ w_off,
                                   const float* __restrict__ b_off,
                                   float* __restrict__ off) {
  __shared__ float S[2][RED * SPITCH];

  int lane = threadIdx.x & 31;
  int wave = threadIdx.x >> 5;

  int bid = blockIdx.x;            // = n*128 + tilePair
  int tp  = bid & 127;
  int n   = bid >> 7;
  int tile  = tp * 2 + wave;       // 0..255
  int h     = tile >> 2;
  int wBase = (tile & 3) << 4;

  int p   = lane & 15;             // pixel column / M row within tile
  int rep = lane >> 4;             // lane half
  int khalf = rep * 2;

  int b = n >> 3, d = n & 7;
  const float* xn = x + ((size_t)b * Cin * Dn + d) * HW;  // + c*Dn*HW
  float* Sw = S[wave];

  // clamped A-row indices per m-tile (uniform across K loop, no divergence)
  int arow[5];
  #pragma unroll
  for (int m = 0; m < 5; ++m) arow[m] = min(m * 16 + p, OFFC - 1);

  v8f acc[5] = {};                 // 5 M-tiles of 16 channels

  for (int cb = 0; cb < 4; ++cb) {
    // ---- build im2col tile for channels [cb*32, cb*32+32) ----
    for (int k = 0; k < KK; ++k) {
      int ky = k / 3, kx = k % 3;
      int yy = h + ky - 1;
      int xx = wBase + p + kx - 1;
      bool valid = (yy >= 0) && (yy < Hh) && (xx >= 0) && (xx < Ww);
      int sidx = yy * Ww + xx;
      for (int cc = rep; cc < CG; cc += 2) {
        int c = cb * CG + cc;
        float v = valid ? xn[(size_t)c * Dn * HW + sidx] : 0.0f;
        Sw[(cc * KK + k) * SPITCH + p] = v;
      }
    }
    __syncthreads();

    // ---- WMMA over this chunk's 288 reduction values ----
    // weight flat index: oc*1152 + (cb*32+cc)*9 + k = oc*1152 + cb*288 + r
    const float* Wg = w_off + cb * RED;
    for (int kk = 0; kk < RED; kk += 4) {
      v2f bf;
      bf.x = Sw[(kk + khalf)     * SPITCH + p];
      bf.y = Sw[(kk + khalf + 1) * SPITCH + p];
      #pragma unroll
      for (int m = 0; m < 5; ++m) {
        const float* wp = Wg + (size_t)arow[m] * (Cin * KK) + kk + khalf;
        v2f a;
        a.x = wp[0];
        a.y = wp[1];
        acc[m] = __builtin_amdgcn_wmma_f32_16x16x4_f32(false, a, false, bf,
                                                       (short)0, acc[m],
                                                       false, false);
      }
    }
    __syncthreads();               // before S is overwritten next chunk
  }

  // ---- writeback + bias; rows >= 72 discarded ----
  int mo = rep * 8;
  #pragma unroll
  for (int m = 0; m < 5; ++m) {
    #pragma unroll
    for (int r = 0; r < 8; ++r) {
      int oc = m * 16 + mo + r;
      if (oc < OFFC) {
        off[((size_t)n * OFFC + oc) * HW + (size_t)h * Ww + wBase + p] =
            acc[m][r] + b_off[oc];
      }
    }
  }
}

// ---------------------------------------------------------------
// Kernel 2: deformable conv via fp32 WMMA (v_wmma_f32_16x16x4_f32).
// Block = 64 threads (2 waves). Each wave owns one 16-pixel tile
// (fixed h, 16 consecutive w) for one (n, group).
//   Stage A: bilinear-sample im2col tile S[288][16] into LDS.
//   Stage B: D[32oc x 16px] = W[32 x 288] * S, K-steps of 4.
// ---------------------------------------------------------------
__global__ void deform_wmma_kernel(const float* __restrict__ x,
                                   const float* __restrict__ off,
                                   const float* __restrict__ w_dc,
                                   const float* __restrict__ b_dc,
                                   float* __restrict__ y) {
  __shared__ float S[2][RED * SPITCH];

  int lane = threadIdx.x & 31;
  int wave = threadIdx.x >> 5;

  int bid = blockIdx.x;            // = ((n*OG + g)*128 + tilePair)
  int tp  = bid & 127;
  int g   = (bid >> 7) & 3;
  int n   = bid >> 9;
  int tile  = tp * 2 + wave;       // 0..255
  int h     = tile >> 2;
  int wBase = (tile & 3) << 4;

  int p   = lane & 15;             // pixel within tile / N column / M row
  int rep = lane >> 4;             // lane half

  int b = n >> 3, d = n & 7;
  // base of this image's group-g channels; channel cg at + cg*Dn*HW
  const float* xg = x + (((size_t)b * Cin + (size_t)g * CG) * Dn + d) * HW;
  float* Sw = S[wave];

  // ---- Stage A: build im2col tile (bilinear, zero padding) ----
  for (int k = 0; k < KK; ++k) {
    int ky = k / 3, kx = k % 3;
    size_t obase = ((size_t)n * OFFC + (size_t)(g * KK + k) * 2) * HW
                 + (size_t)h * Ww + (wBase + p);
    float dy = off[obase];
    float dx = off[obase + HW];
    float py = dy + (float)(ky - 1 + h);
    float px = dx + (float)(kx - 1 + wBase + p);
    float fy0 = floorf(py), fx0 = floorf(px);
    float wy = py - fy0, wx = px - fx0;
    int y0 = (int)fy0, x0 = (int)fx0;
    int y1 = y0 + 1,   x1 = x0 + 1;
    float vy0 = (y0 >= 0 && y0 < Hh) ? 1.0f : 0.0f;
    float vy1 = (y1 >= 0 && y1 < Hh) ? 1.0f : 0.0f;
    float vx0 = (x0 >= 0 && x0 < Ww) ? 1.0f : 0.0f;
    float vx1 = (x1 >= 0 && x1 < Ww) ? 1.0f : 0.0f;
    float w00 = (1.0f - wy) * (1.0f - wx) * vy0 * vx0;
    float w01 = (1.0f - wy) * wx          * vy0 * vx1;
    float w10 = wy          * (1.0f - wx) * vy1 * vx0;
    float w11 = wy          * wx          * vy1 * vx1;
    int iy0 = min(max(y0, 0), Hh - 1), iy1 = min(max(y1, 0), Hh - 1);
    int ix0 = min(max(x0, 0), Ww - 1), ix1 = min(max(x1, 0), Ww - 1);
    int i00 = iy0 * Ww + ix0, i01 = iy0 * Ww + ix1;
    int i10 = iy1 * Ww + ix0, i11 = iy1 * Ww + ix1;
    // lane halves split the 32 channels
    for (int cg = rep; cg < CG; cg += 2) {
      const float* xc = xg + (size_t)cg * Dn * HW;
      float v = w00 * xc[i00] + w01 * xc[i01] + w10 * xc[i10] + w11 * xc[i11];
      Sw[(cg * KK + k) * SPITCH + p] = v;
    }
  }
  __syncthreads();

  // ---- Stage B: fp32 WMMA GEMM ----
  // A 16x4 fp32 layout: lane<16 holds M=lane, K={kk,kk+1}; lane>=16 holds
  // M=lane-16, K={kk+2,kk+3}.  B 4x16: mirrored row striping.
  int khalf = rep * 2;
  const float* Wg = w_dc + (size_t)(g * OPG) * RED;   // [OPG][RED]
  v8f c0 = {};
  v8f c1 = {};
  for (int kk = 0; kk < RED; kk += 4) {
    v2f bf, a0, a1;
    bf.x = Sw[(kk + khalf)     * SPITCH + p];
    bf.y = Sw[(kk + khalf + 1) * SPITCH + p];
    a0.x = Wg[(size_t)p        * RED + kk + khalf];
    a0.y = Wg[(size_t)p        * RED + kk + khalf + 1];
    a1.x = Wg[(size_t)(p + 16) * RED + kk + khalf];
    a1.y = Wg[(size_t)(p + 16) * RED + kk + khalf + 1];
    c0 = __builtin_amdgcn_wmma_f32_16x16x4_f32(false, a0, false, bf,
                                               (short)0, c0, false, false);
    c1 = __builtin_amdgcn_wmma_f32_16x16x4_f32(false, a1, false, bf,
                                               (short)0, c1, false, false);
  }

  // ---- writeback + bias.  C/D layout: VGPR r -> M=r (lanes 0-15) / M=r+8
  int mo = rep * 8;
  float* yp = y + ((size_t)n * OC + (size_t)g * OPG) * HW
                + (size_t)h * Ww + wBase + p;
  #pragma unroll
  for (int r = 0; r < 8; ++r) {
    int o0 = mo + r;                    // mtile 0: channels 0..15 of group
    int o1 = 16 + mo + r;               // mtile 1: channels 16..31 of group
    yp[(size_t)o0 * HW] = c0[r] + b_dc[g * OPG + o0];
    yp[(size_t)o1 * HW] = c1[r] + b_dc[g * OPG + o1];
  }
}

// ---------------------------------------------------------------
// Kernel 3: instance-norm statistics per (n, c) over HW.
// ---------------------------------------------------------------
__global__ void instnorm_stats_kernel(const float* __restrict__ y,
                                      float* __restrict__ stats) {
  __shared__ float ssum[256];
  __shared__ float ssq[256];
  int nc = blockIdx.x;
  int tid = threadIdx.x;
  const float* yp = y + (size_t)nc * HW;
  float s = 0.0f, q = 0.0f;
  for (int i = tid; i < HW; i += 256) {
    float v = yp[i];
    s += v; q += v * v;
  }
  ssum[tid] = s; ssq[tid] = q;
  __syncthreads();
  for (int o = 128; o > 0; o >>= 1) {
    if (tid < o) { ssum[tid] += ssum[tid + o]; ssq[tid] += ssq[tid + o]; }
    __syncthreads();
  }
  if (tid == 0) {
    float mu  = ssum[0] * (1.0f / HW);
    float var = ssq[0] * (1.0f / HW) - mu * mu;
    stats[2 * nc]     = mu;
    stats[2 * nc + 1] = rsqrtf(var + 1e-5f);
  }
}

// ---------------------------------------------------------------
// Kernel 4: normalize + exact GELU + transpose to [B, C, D, H, W].
// ---------------------------------------------------------------
__global__ void finalize_kernel(const float* __restrict__ y,
                                const float* __restrict__ stats,
                                float* __restrict__ out) {
  int idx = blockIdx.x * blockDim.x + threadIdx.x;   // < Nn*OC*HW
  int pix = idx & (HW - 1);
  int t   = idx >> 12;       // = n*OC + c
  int c   = t & 127;
  int n   = t >> 7;
  float mu   = stats[2 * t];
  float rstd = stats[2 * t + 1];
  float v = (y[idx] - mu) * rstd;
  float gl = 0.5f * v * (1.0f + erff(v * 0.70710678118654752f));
  int b = n >> 3, d = n & 7;
  out[(((size_t)b * OC + c) * Dn + d) * HW + pix] = gl;
}

// ---------------------------------------------------------------
extern "C" void kernel_launch(void* const* d_in, const int* in_sizes, int n_in,
                              void* d_out, int out_size, void* d_ws, size_t ws_size,
                              hipStream_t stream) {
  const float* x     = (const float*)d_in[0];
  const float* w_off = (const float*)d_in[1];
  const float* b_off = (const float*)d_in[2];
  const float* w_dc  = (const float*)d_in[3];
  const float* b_dc  = (const float*)d_in[4];
  float* out = (float*)d_out;

  // workspace layout
  float* off_ws = (float*)d_ws;                              // Nn*OFFC*HW
  float* y_ws   = off_ws + (size_t)Nn * OFFC * HW;           // Nn*OC*HW
  float* stats  = y_ws + (size_t)Nn * OC * HW;               // 2*Nn*OC

  // 1) offset conv as WMMA implicit-GEMM: 2 waves/block, 1 tile/wave
  offset_wmma_kernel<<<Nn * 128, 64, 0, stream>>>(x, w_off, b_off, off_ws);
  // 2) deform conv with fp32 WMMA: 2 waves/block, 1 tile/wave
  deform_wmma_kernel<<<Nn * OG * 128, 64, 0, stream>>>(x, off_ws, w_dc, b_dc, y_ws);
  // 3) instance-norm stats: one block per (n, c)
  instnorm_stats_kernel<<<Nn * OC, 256, 0, stream>>>(y_ws, stats);
  // 4) normalize + GELU + transpose
  finalize_kernel<<<(Nn * OC * HW) / 256, 256, 0, stream>>>(y_ws, stats, out);
}